// AttentionEval_3332894622054
// MI455X (gfx1250) — compile-verified
//
#include <hip/hip_runtime.h>
#include <math.h>

// MI455X (gfx1250) single-pass online-softmax attention context kernel.
// Bandwidth-bound: enc is 512MB, one streaming pass @23.3TB/s ~= 22us floor.
// Scores via V_WMMA_F32_16X16X4_F32 (exact fp32); context via fp32 VALU FMA.

typedef float v2f __attribute__((ext_vector_type(2)));
typedef float v8f __attribute__((ext_vector_type(8)));

#define BSZ   64
#define SSZ   2048
#define HSZ   1024
#define TPB   256
#define TILE  16
#define WS_STRIDE 1040   // floats per (b,g) record: [0]=m, [1]=l, [16..16+1024)=ctx

__global__ __launch_bounds__(TPB)
void attn_pass1(const float* __restrict__ enc, const float* __restrict__ hn,
                float* __restrict__ out, float* __restrict__ ws,
                int groups, int rows_per_group)
{
    __shared__ float q_lds[HSZ];
    __shared__ float s_part[8][16];
    __shared__ float s_scores[16];

    const int tid  = threadIdx.x;
    const int lane = tid & 31;
    const int wv   = tid >> 5;                 // wave id 0..7 (wave32)
    const int b    = blockIdx.x / groups;
    const int g    = blockIdx.x % groups;

    const float* encB = enc + (size_t)b * SSZ * HSZ;

    // Stage q (h_n row) in LDS, coalesced.
    for (int i = tid; i < HSZ; i += TPB) q_lds[i] = hn[(size_t)b * HSZ + i];
    __syncthreads();

    // Preload A fragments: A[m][k] = q[k] broadcast over M.
    // A layout (16x4 f32): lane l -> M=l%16; VGPR0 = {K0 | K2}, VGPR1 = {K1 | K3}.
    const int koff = (lane < 16) ? 0 : 2;
    v2f qa[32];
#pragma unroll
    for (int i = 0; i < 32; ++i) {
        const int k = wv * 128 + i * 4;
        qa[i] = *(const v2f*)&q_lds[k + koff];
    }

    const int c0 = tid * 4;                    // context columns owned by this thread
    float ctx0 = 0.f, ctx1 = 0.f, ctx2 = 0.f, ctx3 = 0.f;
    float m_run = -INFINITY, l_run = 0.f;

    const int row_base = g * rows_per_group;
    const int ntiles   = rows_per_group / TILE;

    for (int t = 0; t < ntiles; ++t) {
        const int r0 = row_base + t * TILE;

        // ---- Phase B: 16 row-scores via WMMA. B[k][n] = enc[r0+n][k] (4x16),
        //      same {K0|K2}/{K1|K3} half-wave pattern as A, N = lane%16.
        const float* bp = encB + (size_t)(r0 + (lane & 15)) * HSZ + koff + wv * 128;
        v8f ce = {}, co = {};
#pragma unroll
        for (int i = 0; i < 32; i += 2) {
            v2f b0 = *(const v2f*)(bp + i * 4);
            v2f b1 = *(const v2f*)(bp + (i + 1) * 4);
            ce = __builtin_amdgcn_wmma_f32_16x16x4_f32(false, qa[i],     false, b0,
                                                       (short)0, ce, false, false);
            co = __builtin_amdgcn_wmma_f32_16x16x4_f32(false, qa[i + 1], false, b1,
                                                       (short)0, co, false, false);
        }
        v8f cs = ce + co;                       // D rows identical; row M=0 = scores
        if (lane < 16) s_part[wv][lane] = cs[0];
        __syncthreads();
        if (tid < 16) {
            float acc = 0.f;
#pragma unroll
            for (int w2 = 0; w2 < 8; ++w2) acc += s_part[w2][tid];
            s_scores[tid] = acc;
        }
        __syncthreads();

        // ---- Phase C: online softmax update + context GEMV (exact fp32 FMA).
        float s[TILE];
        float tmax = -INFINITY;
#pragma unroll
        for (int r = 0; r < TILE; ++r) { s[r] = s_scores[r]; tmax = fmaxf(tmax, s[r]); }
        const float m_new = fmaxf(m_run, tmax);
        const float scale = __expf(m_run - m_new);   // exp(-inf)=0 on first tile
        l_run *= scale; ctx0 *= scale; ctx1 *= scale; ctx2 *= scale; ctx3 *= scale;

        const float* cp = encB + (size_t)r0 * HSZ + c0;   // L2-hot (just streamed)
#pragma unroll
        for (int r = 0; r < TILE; ++r) {
            const float p = __expf(s[r] - m_new);
            l_run += p;
            const float4 e = *(const float4*)(cp + (size_t)r * HSZ);
            ctx0 += p * e.x; ctx1 += p * e.y; ctx2 += p * e.z; ctx3 += p * e.w;
        }
        m_run = m_new;

        // Prefetch next tile into caches for the next WMMA phase.
        if ((t + 1 < ntiles) && ((tid & 3) == 0)) {
            const float* pf = encB + (size_t)(r0 + TILE) * HSZ + c0;
#pragma unroll
            for (int r = 0; r < TILE; ++r)
                __builtin_prefetch(pf + (size_t)r * HSZ, 0, 1);
        }
        __syncthreads();
    }

    if (groups == 1) {
        const float inv = 1.f / l_run;
        float* ob = out + (size_t)b * (2 * HSZ);
        ob[HSZ + c0 + 0] = ctx0 * inv;
        ob[HSZ + c0 + 1] = ctx1 * inv;
        ob[HSZ + c0 + 2] = ctx2 * inv;
        ob[HSZ + c0 + 3] = ctx3 * inv;
        for (int i = tid; i < HSZ; i += TPB) ob[i] = q_lds[i];   // dec_output copy
    } else {
        float* wsb = ws + (size_t)(b * groups + g) * WS_STRIDE;
        if (tid == 0) { wsb[0] = m_run; wsb[1] = l_run; }
        wsb[16 + c0 + 0] = ctx0;
        wsb[16 + c0 + 1] = ctx1;
        wsb[16 + c0 + 2] = ctx2;
        wsb[16 + c0 + 3] = ctx3;
    }
}

__global__ __launch_bounds__(TPB)
void attn_pass2(const float* __restrict__ hn, const float* __restrict__ ws,
                float* __restrict__ out, int groups)
{
    const int tid = threadIdx.x;
    const int b   = blockIdx.x;

    float M = -INFINITY;
    for (int g = 0; g < groups; ++g)
        M = fmaxf(M, ws[(size_t)(b * groups + g) * WS_STRIDE]);

    float L = 0.f;
    float wgt[8];
    for (int g = 0; g < groups; ++g) {
        const float* wsb = ws + (size_t)(b * groups + g) * WS_STRIDE;
        const float w = __expf(wsb[0] - M);
        wgt[g] = w;
        L += w * wsb[1];
    }

    const int c0 = tid * 4;
    float a0 = 0.f, a1 = 0.f, a2 = 0.f, a3 = 0.f;
    for (int g = 0; g < groups; ++g) {
        const float* wsb = ws + (size_t)(b * groups + g) * WS_STRIDE + 16 + c0;
        const float4 v = *(const float4*)wsb;
        a0 += wgt[g] * v.x; a1 += wgt[g] * v.y;
        a2 += wgt[g] * v.z; a3 += wgt[g] * v.w;
    }
    const float inv = 1.f / L;
    float* ob = out + (size_t)b * (2 * HSZ);
    ob[HSZ + c0 + 0] = a0 * inv;
    ob[HSZ + c0 + 1] = a1 * inv;
    ob[HSZ + c0 + 2] = a2 * inv;
    ob[HSZ + c0 + 3] = a3 * inv;
    for (int i = tid; i < HSZ; i += TPB) ob[i] = hn[(size_t)b * HSZ + i];
}

extern "C" void kernel_launch(void* const* d_in, const int* in_sizes, int n_in,
                              void* d_out, int out_size, void* d_ws, size_t ws_size,
                              hipStream_t stream) {
    const float* enc = (const float*)d_in[0];   // (64, 2048, 1024) f32
    const float* hn  = (const float*)d_in[1];   // (64, 1024) f32
    float* out = (float*)d_out;                 // (64, 1, 2048) f32
    float* ws  = (float*)d_ws;

    // Split S across groups for occupancy (512 blocks); fall back if ws too small.
    int groups = 1;
    const int cand[3] = {8, 4, 2};
    for (int i = 0; i < 3; ++i) {
        if ((size_t)BSZ * cand[i] * WS_STRIDE * sizeof(float) <= ws_size) {
            groups = cand[i];
            break;
        }
    }
    const int rpg = SSZ / groups;

    attn_pass1<<<dim3(BSZ * groups), dim3(TPB), 0, stream>>>(enc, hn, out, ws, groups, rpg);
    if (groups > 1)
        attn_pass2<<<dim3(BSZ), dim3(TPB), 0, stream>>>(hn, ws, out, groups);
}